// TreeMultiRandom_40707700032067
// MI455X (gfx1250) — compile-verified
//
#include <hip/hip_runtime.h>

// ---------------------------------------------------------------------------
// KNN inpainting on gfx1250 (exact-arithmetic formulation):
//   WMMA quadratic form with RAW integer coords:
//     [-2ur,-2uc,ur^2+uc^2,1] . [fr,fc,1,fr^2+fc^2] = (dr)^2+(dc)^2  (exact:
//   every term/partial sum is an integer < 2^24 -> fp32 exact).
//   Pack key = (id2<<15)|colIdx -> u32; top-3 = pure v_min_u32/v_max_u32
//   network (no branches, matches reference tie-breaks exactly).
//   Sweep loop: software-pipelined coord load + unroll 2 to kill the
//   loop-edge register-permutation movs.
// ---------------------------------------------------------------------------

typedef float v2f __attribute__((ext_vector_type(2)));
typedef float v8f __attribute__((ext_vector_type(8)));

#define WAVES 4  // waves per block

__device__ __forceinline__ unsigned umn(unsigned a, unsigned b) { return a < b ? a : b; }
__device__ __forceinline__ unsigned umx(unsigned a, unsigned b) { return a > b ? a : b; }

// insert key k into sorted triple s0<=s1<=s2 (keep 3 smallest) — 5 min/max ops
__device__ __forceinline__ void ins3u(unsigned k,
                                      unsigned& s0, unsigned& s1, unsigned& s2) {
  unsigned t0 = umn(s0, k);
  unsigned m0 = umx(s0, k);
  unsigned t1 = umn(s1, m0);
  unsigned m1 = umx(s1, m0);
  unsigned t2 = umn(s2, m1);
  s0 = t0; s1 = t1; s2 = t2;
}

__global__ void __launch_bounds__(WAVES * 32)
knn_fill_kernel(const float* __restrict__ img,      // 8 x 65536
                const int2*  __restrict__ filled,   // nf x (row,col)
                const int2*  __restrict__ unfilled, // nu x (row,col)
                int nf, int cls,
                float* __restrict__ out)            // 8 x 4 x 65536
{
  __shared__ unsigned sk[WAVES][16][3];

  const int  lane = threadIdx.x & 31;
  const int  wave = threadIdx.x >> 5;
  const int  tile = blockIdx.x * WAVES + wave;   // 16-row tile
  const int  row0 = tile * 16;
  const int  m    = lane & 15;
  const bool hi   = lane >= 16;

  // ---- A fragment: row M = row0+m, vector [-2ur, -2uc, ur^2+uc^2, 1] ----
  // fp32 16x4 A layout: lanes 0-15 hold K=0,1; lanes 16-31 hold K=2,3.
  int2  u  = unfilled[row0 + m];
  float ur = (float)u.x, uc = (float)u.y;
  v2f A;
  A.x = hi ? (ur * ur + uc * uc) : (-2.0f * ur);
  A.y = hi ? 1.0f                : (-2.0f * uc);

  unsigned k0[8], k1[8], k2[8];
#pragma unroll
  for (int r = 0; r < 8; ++r) { k0[r] = k1[r] = k2[r] = 0xffffffffu; }

  // ---- sweep filled points, 16 columns per WMMA ----
  // Software-pipelined: fetch next column-tile's coords while the current
  // WMMA + selection network executes. nf/16 is even for all classes.
  const int2* fp   = filled + m;
  int2        fcur = fp[0];
#pragma unroll 2
  for (int c = 0; c < nf; c += 16) {
    const int nxt   = (c + 16 < nf) ? (c + 16) : 0;
    int2      fnext = fp[nxt];                 // in flight during this tile

    // B fragment: col N = c+m, vector [fr, fc, 1, fr^2+fc^2]
    float fr = (float)fcur.x, fc = (float)fcur.y;
    v2f B;
    B.x = hi ? 1.0f                : fr;
    B.y = hi ? (fr * fr + fc * fc) : fc;

    v8f C = {0.f, 0.f, 0.f, 0.f, 0.f, 0.f, 0.f, 0.f};
    C = __builtin_amdgcn_wmma_f32_16x16x4_f32(
            /*neg_a=*/false, A, /*neg_b=*/false, B,
            /*c_mod=*/(short)0, C, /*reuse_a=*/false, /*reuse_b=*/false);

    const unsigned col = (unsigned)(c + m);    // N = lane%16 in C/D layout
#pragma unroll
    for (int r = 0; r < 8; ++r) {              // reg r: row r (lo) / 8+r (hi)
      unsigned key = ((unsigned)C[r] << 15) | col;  // exact int d2 in [1, 2^17)
      ins3u(key, k0[r], k1[r], k2[r]);
    }
    fcur = fnext;
  }

  // ---- merge top-3 across the 16 lanes of each half-wave ----
#pragma unroll
  for (int off = 1; off < 16; off <<= 1) {
#pragma unroll
    for (int r = 0; r < 8; ++r) {
      unsigned b0 = (unsigned)__shfl_xor((int)k0[r], off, 16);
      unsigned b1 = (unsigned)__shfl_xor((int)k1[r], off, 16);
      unsigned b2 = (unsigned)__shfl_xor((int)k2[r], off, 16);
      ins3u(b0, k0[r], k1[r], k2[r]);
      ins3u(b1, k0[r], k1[r], k2[r]);
      ins3u(b2, k0[r], k1[r], k2[r]);
    }
  }

  // ---- park merged results in LDS (one lane per half-wave) ----
  if (m == 0) {
    const int base = hi ? 8 : 0;
#pragma unroll
    for (int r = 0; r < 8; ++r) {
      sk[wave][base + r][0] = k0[r];
      sk[wave][base + r][1] = k1[r];
      sk[wave][base + r][2] = k2[r];
    }
  }
  __syncthreads();

  // ---- IDW blend: lane handles row m, batches [hi*4, hi*4+4) ----
  unsigned key0 = sk[wave][m][0];
  unsigned key1 = sk[wave][m][1];
  unsigned key2 = sk[wave][m][2];

  // dist_norm = sqrt(id2)/128  ->  w = 1/dist_norm = 128/sqrt(id2)
  float w0 = 128.0f / sqrtf((float)(key0 >> 15));
  float w1 = 128.0f / sqrtf((float)(key1 >> 15));
  float w2 = 128.0f / sqrtf((float)(key2 >> 15));
  float wsum = w0 + w1 + w2;

  int2 p0 = filled[key0 & 0x7fffu];
  int2 p1 = filled[key1 & 0x7fffu];
  int2 p2 = filled[key2 & 0x7fffu];
  int  o0 = p0.x * 256 + p0.y;
  int  o1 = p1.x * 256 + p1.y;
  int  o2 = p2.x * 256 + p2.y;

  int2 up     = unfilled[row0 + m];
  int  up_off = up.x * 256 + up.y;
  int  b0     = hi ? 4 : 0;

#pragma unroll
  for (int k = 0; k < 4; ++k) {
    int   b = b0 + k;
    float v = w0 * img[b * 65536 + o0] +
              w1 * img[b * 65536 + o1] +
              w2 * img[b * 65536 + o2];
    out[(b * 4 + cls) * 65536 + up_off] = v / wsum;
  }
}

// Pass-through of filled pixels for this class.
__global__ void fill_pass_kernel(const float* __restrict__ img,
                                 const int2* __restrict__ filled,
                                 int nf, int cls,
                                 float* __restrict__ out)
{
  int t = blockIdx.x * blockDim.x + threadIdx.x;
  if (t >= nf) return;
  int2 p   = filled[t];
  int  off = p.x * 256 + p.y;
#pragma unroll
  for (int b = 0; b < 8; ++b)
    out[(b * 4 + cls) * 65536 + off] = img[b * 65536 + off];
}

extern "C" void kernel_launch(void* const* d_in, const int* in_sizes, int n_in,
                              void* d_out, int out_size, void* d_ws, size_t ws_size,
                              hipStream_t stream)
{
  const float* img = (const float*)d_in[0];   // coded (8,1,256,256) f32
  float*       out = (float*)d_out;           // (8,4,256,256) f32

  for (int cls = 0; cls < 4; ++cls) {
    const int2* filled   = (const int2*)d_in[2 + 2 * cls];
    const int2* unfilled = (const int2*)d_in[3 + 2 * cls];
    const int   nf = in_sizes[2 + 2 * cls] / 2;
    const int   nu = in_sizes[3 + 2 * cls] / 2;

    fill_pass_kernel<<<(nf + 255) / 256, 256, 0, stream>>>(img, filled, nf, cls, out);

    // nu is a multiple of 16*WAVES for all four classes (3584/3072/3584/2048 tiles)
    const int tiles = nu / 16;
    knn_fill_kernel<<<tiles / WAVES, WAVES * 32, 0, stream>>>(
        img, filled, unfilled, nf, cls, out);
  }
}